// GCNLayer_67559835566264
// MI455X (gfx1250) — compile-verified
//
#include <hip/hip_runtime.h>
#include <hip/hip_bf16.h>

typedef float v2f __attribute__((ext_vector_type(2)));
typedef float v8f __attribute__((ext_vector_type(8)));

#define D_FEAT 128

// ---------------------------------------------------------------------------
// Kernel 1: segmented SpMM.  edge_row is sorted, so each node row's edges form
// a contiguous run [start,end).  One wave32 per row; every lane redundantly
// binary-searches the run bounds (~20 SALU/VALU steps, cheap), then lane t
// accumulates features [4t, 4t+4) as a float4 -> global_load_b128 gathers,
// 512B coalesced per edge.  x (25.6 MB) is L2-resident (192 MB L2), so the
// gather stream is an L2-bandwidth problem, not an HBM one.  No atomics, fully
// deterministic, and empty rows naturally write zeros (no memset needed).
// ---------------------------------------------------------------------------
__global__ __launch_bounds__(32) void gcn_spmm_rowwave(
    const float* __restrict__ x,
    const int*   __restrict__ erow,
    const int*   __restrict__ ecol,
    const float* __restrict__ eval,
    float*       __restrict__ agg,
    int n_edges)
{
    const int row = blockIdx.x;

    // lower_bound(row)
    int lo = 0, hi = n_edges;
    while (lo < hi) { int mid = (lo + hi) >> 1; if (erow[mid] <  row) lo = mid + 1; else hi = mid; }
    const int start = lo;
    // upper_bound(row), searching only [start, n_edges)
    hi = n_edges;
    while (lo < hi) { int mid = (lo + hi) >> 1; if (erow[mid] <= row) lo = mid + 1; else hi = mid; }
    const int end = lo;

    const int t = threadIdx.x;                  // 0..31, 4 features each
    const float4* __restrict__ x4 = (const float4*)x;
    float4 acc = make_float4(0.f, 0.f, 0.f, 0.f);

    for (int e = start; e < end; ++e) {
        const float v = eval[e];                // wave-uniform broadcast load
        const int   c = ecol[e];
        const float4 xv = x4[c * (D_FEAT / 4) + t];
        acc.x += v * xv.x;  acc.y += v * xv.y;
        acc.z += v * xv.z;  acc.w += v * xv.w;
    }
    ((float4*)agg)[row * (D_FEAT / 4) + t] = acc;
}

// ---------------------------------------------------------------------------
// Kernel 2: out = agg @ W^T + b  via V_WMMA_F32_16X16X4_F32 (full fp32 — the
// GEMM is HBM/L2 bound at 51 MB for 1.64 GFLOP, so low-precision WMMA buys
// nothing; fp32 WMMA keeps exact reference numerics on the WMMA pipe).
//
// Per ISA 7.12.2 (f32, wave32):
//   A 16x4 : lane L -> M = L%16, K = k + (L/16)*2 + {0,1}  (2 VGPRs = v2f)
//   B 4x16 : symmetric, lane L -> N = L%16; B[k][n] = W[n][k] so we read
//            float2 pairs straight out of row W[n][..]
//   C/D    : v8f; lane L holds N = L%16, M = (L/16)*8 + r  -> bias splat.
// One 256-thread block = 8 waves = one 16x128 output slab; wave w owns
// N-tile w.  K=128 -> 32 chained WMMAs, accumulator lives in VGPRs.
// 50000 % 16 == 0 and 128 == 8*16, so no edge guards.
// ---------------------------------------------------------------------------
__global__ __launch_bounds__(256) void gcn_gemm_wmma_f32(
    const float* __restrict__ agg,
    const float* __restrict__ W,
    const float* __restrict__ bias,
    float*       __restrict__ out)
{
    const int lane  = threadIdx.x & 31;
    const int wave  = threadIdx.x >> 5;          // 0..7 : N tile
    const int row0  = blockIdx.x * 16;           // 16 M-rows per block
    const int nIdx  = wave * 16 + (lane & 15);   // this lane's output column
    const int mIdx  = row0 + (lane & 15);        // this lane's A row
    const int koff  = (lane >> 4) * 2;           // 0 or 2

    v8f c;
    const float bv = bias[nIdx];
    #pragma unroll
    for (int r = 0; r < 8; ++r) c[r] = bv;

    const v2f* __restrict__ Arow = (const v2f*)(agg + mIdx * D_FEAT);
    const v2f* __restrict__ Brow = (const v2f*)(W   + nIdx * D_FEAT);

    #pragma unroll 4
    for (int k = 0; k < D_FEAT; k += 4) {
        v2f a = Arow[(k + koff) >> 1];
        v2f b = Brow[(k + koff) >> 1];
        // (neg_a, A, neg_b, B, c_mod, C, reuse_a, reuse_b)
        c = __builtin_amdgcn_wmma_f32_16x16x4_f32(
                false, a, false, b, (short)0, c, false, false);
    }

    const int mbase = row0 + (lane >> 4) * 8;
    const int col   = wave * 16 + (lane & 15);
    #pragma unroll
    for (int r = 0; r < 8; ++r)
        out[(mbase + r) * D_FEAT + col] = c[r];
}

// ---------------------------------------------------------------------------
extern "C" void kernel_launch(void* const* d_in, const int* in_sizes, int n_in,
                              void* d_out, int out_size, void* d_ws, size_t ws_size,
                              hipStream_t stream)
{
    const float* x    = (const float*)d_in[0];
    const int*   erow = (const int*)  d_in[1];
    const int*   ecol = (const int*)  d_in[2];
    const float* eval = (const float*)d_in[3];
    const float* W    = (const float*)d_in[4];
    const float* bias = (const float*)d_in[5];
    float*       out  = (float*)d_out;

    const int n_nodes = in_sizes[0] / D_FEAT;   // 50000
    const int n_edges = in_sizes[1];            // 800000

    float* agg = (float*)d_ws;                  // n_nodes * 128 * 4B = 25.6 MB

    gcn_spmm_rowwave<<<n_nodes, 32, 0, stream>>>(x, erow, ecol, eval, agg, n_edges);
    gcn_gemm_wmma_f32<<<n_nodes / 16, 256, 0, stream>>>(agg, W, bias, out);
}